// TemporalFoldTranspose1d_86947317940455
// MI455X (gfx1250) — compile-verified
//
#include <hip/hip_runtime.h>
#include <cstdint>
#include <cstddef>

// Problem constants (from reference: B=8, C=256, K=16, L=2048, stride=8)
#define K_      16
#define STRIDE_ 8
#define B_      8
#define C_      256
#define L_      2048
#define OUTL_   (L_ * STRIDE_)     // 16384
#define CHUNK_  256                // output t0-tiles per workgroup
#define NCHUNK_ (L_ / CHUNK_)      // 8 chunks per (b,c) row
#define PITCH_  264                // LDS floats per k-row: CHUNK_+2 halo, padded to 66 float4s

// ---- CDNA5 async global->LDS copy (ASYNCcnt-tracked) -----------------------
typedef int v4i_ __attribute__((__vector_size__(4 * sizeof(int))));
typedef __attribute__((address_space(1))) v4i_* glb_v4i_ptr;
typedef __attribute__((address_space(3))) v4i_* lds_v4i_ptr;
typedef float f4_ __attribute__((ext_vector_type(4)));

__device__ __forceinline__ void async_copy_b128(const float* g, float* l) {
#if __has_builtin(__builtin_amdgcn_global_load_async_to_lds_b128)
    __builtin_amdgcn_global_load_async_to_lds_b128(
        (glb_v4i_ptr)g, (lds_v4i_ptr)l, 0, 0);
#else
    unsigned loff = (unsigned)(uintptr_t)(__attribute__((address_space(3))) void*)l;
    asm volatile("global_load_async_to_lds_b128 %0, %1, off"
                 :: "v"(loff), "v"(g) : "memory");
#endif
}

__device__ __forceinline__ void wait_async_zero() {
#if __has_builtin(__builtin_amdgcn_s_wait_asynccnt)
    __builtin_amdgcn_s_wait_asynccnt(0);
#else
    asm volatile("s_wait_asynccnt 0x0" ::: "memory");
#endif
}

// ---------------------------------------------------------------------------
// One block handles one (b,c) row and a chunk of 256 output tiles (2048 floats).
// Stage 1: async-stream 16 k-rows x 264 floats of x into LDS (coalesced b128).
// Stage 2: each thread builds one 8-wide output tile from LDS, stores 2x NT float4.
//   out[8*t0 + 0]    = x[k=15][t0]    + x[k=7][t0+1]
//   out[8*t0 + r>=1] = x[k=r+7][t0+1] + x[k=r-1][t0+2]     (t >= L => 0)
// ---------------------------------------------------------------------------
__global__ __launch_bounds__(256) void TemporalFoldTranspose1d_kernel(
        const float* __restrict__ x, float* __restrict__ out) {
    __shared__ __align__(16) float sh[K_ * PITCH_];

    const int bid    = blockIdx.x;
    const int row    = bid >> 3;                 // b*C + c   (NCHUNK_ == 8)
    const int t0base = (bid & (NCHUNK_ - 1)) * CHUNK_;
    const int tid    = threadIdx.x;

    // ---- Stage 1: global -> LDS via async b128 copies (fully unrolled) ----
    {
        const int j = tid & 15;                  // float4 column within row
        const int k = tid >> 4;                  // which of the 16 k-rows
        const float* xrow = x + (((size_t)row * K_ + (size_t)k) << 11); // * L_
        float* lrow = &sh[k * PITCH_];
#pragma unroll
        for (int u = 0; u < 4; ++u) {            // 4 unconditional copies/thread
            const int idx = j + u * 16;          // 0..63 -> floats 0..255
            const int t4  = t0base + (idx << 2);
            const int gt  = (t4 < L_) ? t4 : (L_ - 4);  // clamp; masked in stage 2
            async_copy_b128(xrow + gt, lrow + (idx << 2));
        }
        if (tid < 32) {                          // halo: floats 256..263, wave 0 only
            const int kh  = tid >> 1;
            const int idx = 64 + (tid & 1);
            const int t4  = t0base + (idx << 2);
            const int gt  = (t4 < L_) ? t4 : (L_ - 4);
            async_copy_b128(x + (((size_t)row * K_ + (size_t)kh) << 11) + gt,
                            &sh[kh * PITCH_ + (idx << 2)]);
        }
    }
    wait_async_zero();      // my wave's async LDS writes are complete
    __syncthreads();        // everyone's writes visible to everyone

    // ---- Stage 2: overlap-add gather from LDS, contiguous NT float4 stores ----
    const int  i  = tid;
    const int  t0 = t0base + i;
    const bool v1 = (t0 + 1) < L_;
    const bool v2 = (t0 + 2) < L_;

    float r[8];
    r[0] = sh[15 * PITCH_ + i] + (v1 ? sh[7 * PITCH_ + i + 1] : 0.0f);
#pragma unroll
    for (int q = 1; q < 8; ++q) {
        float a = v1 ? sh[(7 + q) * PITCH_ + i + 1] : 0.0f;
        float b = v2 ? sh[(q - 1) * PITCH_ + i + 2] : 0.0f;
        r[q] = a + b;
    }

    float* op = out + (size_t)row * OUTL_ + ((size_t)t0 << 3);
    f4_ lo = { r[0], r[1], r[2], r[3] };
    f4_ hi = { r[4], r[5], r[6], r[7] };
    __builtin_nontemporal_store(lo, (f4_*)op);        // streamed, written once
    __builtin_nontemporal_store(hi, (f4_*)(op + 4));
}

extern "C" void kernel_launch(void* const* d_in, const int* in_sizes, int n_in,
                              void* d_out, int out_size, void* d_ws, size_t ws_size,
                              hipStream_t stream) {
    const float* x  = (const float*)d_in[0];
    float*      out = (float*)d_out;
    (void)in_sizes; (void)n_in; (void)out_size; (void)d_ws; (void)ws_size;

    dim3 grid(B_ * C_ * NCHUNK_);   // 8*256*8 = 16384 blocks
    dim3 block(256);                // 8 wave32s
    TemporalFoldTranspose1d_kernel<<<grid, block, 0, stream>>>(x, out);
}